// MultislicePropagation_17420387352609
// MI455X (gfx1250) — compile-verified
//
#include <hip/hip_runtime.h>
#include <math.h>

// ---------------------------------------------------------------------------
// Multislice angular-spectrum propagation on MI455X (gfx1250).
// FFTs built from V_WMMA_F32_16X16X4_F32: 512 = 16 x 16 x 2 mixed radix,
// radix-16 stages as complex 16x16 DFT-matrix x 16x32 data matmuls on the
// fp32 WMMA unit. One wave32 per 512-point FFT line (EXEC all ones).
// All LDS data (A twiddle fragments AND B data) stored in fragment-ready
// pair layout: every WMMA operand is one aligned ds_load_b64 straight into
// an even VGPR pair. No VALU repacking between LDS and the WMMA unit.
// ---------------------------------------------------------------------------

typedef __attribute__((ext_vector_type(2))) float v2f;
typedef __attribute__((ext_vector_type(8))) float v8f;

#define TWO_PI 6.28318530717958647692f
#define N1D    512
#define NZ     256
#define CPITCH 520   // column pitch (floats): not a multiple of 64 -> bank spread

// Twiddle table sizes (2 directions: 0 = forward e^{-i..}, 1 = inverse e^{+i..})
#define TABA_N   256   // float2 x3 planes: [d][lane][chunk]  2*32*4
#define TABT1_N  1024  // float2: [d][k1*32+n2]               2*512
#define TABW32_N 32    // float2: [d][j1]                     2*16

__device__ __forceinline__ v8f wmma4(v2f a, v2f b, v8f c) {
  // D = A(16x4, f32) * B(4x16, f32) + C(16x16, f32)
  return __builtin_amdgcn_wmma_f32_16x16x4_f32(
      /*neg_a=*/false, a, /*neg_b=*/false, b,
      /*c_mod=*/(short)0, c, /*reuse_a=*/false, /*reuse_b=*/false);
}

// Fragment-pair permutation: element n (n1 = n>>5 in [0,16), n2 = n&31) lives
// at p(n) = (n1>>1)*64 + n2*2 + (n1&1). The stage-1 pair (kb, kb+1), kb even,
// is then adjacent and 8-byte aligned.
__device__ __forceinline__ int permi(int n) {
  return ((n >> 6) << 6) | ((n & 31) << 1) | ((n >> 5) & 1);
}

// One 512-point complex FFT executed by a single wave32.
//   bR/bI: LDS planes, 512 contiguous floats, input in PERMUTED layout;
//          output written natural (out_perm=0) or permuted (out_perm=1)
//   sR/sI: LDS scratch planes, 512 contiguous (private to this wave)
//   sAr/sAi/sAin: fragment-ready DFT-16 matrix pairs (imag-negated plane too)
//   sT1/sW32: twiddle tables for the chosen direction
//   scale: applied once at the end (1/512 for inverse)
__device__ __forceinline__ void fft512_wave(
    float* __restrict__ bR, float* __restrict__ bI,
    float* __restrict__ sR, float* __restrict__ sI,
    const float2* __restrict__ sAr, const float2* __restrict__ sAi,
    const float2* __restrict__ sAin, const float2* __restrict__ sT1,
    const float2* __restrict__ sW32, float scale, int out_perm, unsigned lane) {
  const unsigned lhalf = lane >> 4;   // 0 or 1
  const unsigned llow  = lane & 15;

  // ---- A fragments: DFT-16 matrix, fragment-ready pairs per (lane, K-chunk).
  // ISA layout, 32-bit A 16x4: lane holds row M=lane&15; K pair = 4c+2*(lane>>4)+{0,1}
  v2f ar[4], ai[4], ain[4];
#pragma unroll
  for (int c = 0; c < 4; ++c) {
    ar[c]  = *(const v2f*)&sAr[(lane << 2) + c];   // one ds_load_b64 each
    ai[c]  = *(const v2f*)&sAi[(lane << 2) + c];
    ain[c] = *(const v2f*)&sAin[(lane << 2) + c];
  }

  // ---- Stage 1: Y[k1,n2] = sum_n1 W16[k1,n1] * x[32*n1+n2], n2 in [0,32)
  //      then twiddle by W512^(k1*n2); scratch in stage-2 fragment layout.
#pragma unroll
  for (int t = 0; t < 2; ++t) {
    const int n2 = t * 16 + (int)llow;   // this lane's data column (B: N = lane&15)
    v8f accr = {}; v8f acci = {};
#pragma unroll
    for (int c = 0; c < 4; ++c) {
      int kb = 4 * c + 2 * (int)lhalf;   // even: pair adjacent in perm layout
      int base = ((kb >> 1) << 6) + (n2 << 1);
      v2f br = *(const v2f*)&bR[base];   // one ds_load_b64 -> aligned pair
      v2f bi = *(const v2f*)&bI[base];
      accr = wmma4(ar[c],  br, accr);    // +Wr*Xr
      accr = wmma4(ain[c], bi, accr);    // -Wi*Xi
      acci = wmma4(ar[c],  bi, acci);    // +Wr*Xi
      acci = wmma4(ai[c],  br, acci);    // +Wi*Xr
    }
    // C layout: lane N = lane&15, VGPR v -> M = v + 8*(lane>>4)
    const int m1 = n2 >> 1, m2 = n2 & 1;
#pragma unroll
    for (int v = 0; v < 8; ++v) {
      int k1 = v + 8 * (int)lhalf;
      float2 tw = sT1[k1 * 32 + n2];
      float yr = accr[v], yi = acci[v];
      int q = k1 * 32 + m2 * 16 + m1;    // stage-2 pairs (m1, m1+1) adjacent
      sR[q] = yr * tw.x - yi * tw.y;
      sI[q] = yr * tw.y + yi * tw.x;
    }
  }
  // same-wave LDS ops are in-order (DScnt): stage-2 reads see stage-1 writes.

  // ---- Stage 2: DFT-32 per k1 row, itself 16x2:
  //      Z[m1,(k1,m2)] = Y[k1, 2*m1+m2]; D = W16 * Z; twiddle W32^(m2*j1);
  //      radix-2 over m2 in registers; natural-order output k = k1+16*j1+256*j2.
  v8f r0 = {}, i0v = {}, r1 = {}, i1v = {};
#pragma unroll
  for (int t = 0; t < 2; ++t) {          // t == m2
    v8f accr = {}; v8f acci = {};
    const int k1 = (int)llow;            // column within tile
#pragma unroll
    for (int c = 0; c < 4; ++c) {
      int kb = 4 * c + 2 * (int)lhalf;   // even: pair adjacent in scratch layout
      int base = k1 * 32 + t * 16 + kb;
      v2f br = *(const v2f*)&sR[base];
      v2f bi = *(const v2f*)&sI[base];
      accr = wmma4(ar[c],  br, accr);
      accr = wmma4(ain[c], bi, accr);
      acci = wmma4(ar[c],  bi, acci);
      acci = wmma4(ai[c],  br, acci);
    }
    if (t == 0) { r0 = accr; i0v = acci; } else { r1 = accr; i1v = acci; }
  }
#pragma unroll
  for (int v = 0; v < 8; ++v) {
    int j1 = v + 8 * (int)lhalf;
    int k1 = (int)llow;
    float2 tw = sW32[j1];                 // W32^(m2=1 * j1)
    float d1r = r1[v] * tw.x - i1v[v] * tw.y;
    float d1i = r1[v] * tw.y + i1v[v] * tw.x;
    float d0r = r0[v], d0i = i0v[v];
    int ka = k1 + 16 * j1;
    int kb2 = ka + 256;
    int ia = out_perm ? permi(ka)  : ka;
    int ib = out_perm ? permi(kb2) : kb2;
    bR[ia] = (d0r + d1r) * scale;  bI[ia] = (d0i + d1i) * scale;
    bR[ib] = (d0r - d1r) * scale;  bI[ib] = (d0i - d1i) * scale;
  }
}

// Stage twiddle tables from global (L2-resident) into LDS; call before sync.
__device__ __forceinline__ void load_tables(
    float2* sAr, float2* sAi, float2* sAin, float2* sT1, float2* sW32,
    const float2* tabAr, const float2* tabAi, const float2* tabAin,
    const float2* tabT1, const float2* tabW32) {
  for (int i = (int)threadIdx.x; i < TABA_N; i += 256) {
    sAr[i] = tabAr[i]; sAi[i] = tabAi[i]; sAin[i] = tabAin[i];
  }
  for (int i = (int)threadIdx.x; i < TABT1_N; i += 256) sT1[i] = tabT1[i];
  if (threadIdx.x < TABW32_N) sW32[threadIdx.x] = tabW32[threadIdx.x];
}

// ---------------------------------------------------------------------------
// One-time twiddle tables (tiny; computed per call for determinism).
__global__ __launch_bounds__(256) void msp_init_tw(
    float2* __restrict__ tabAr, float2* __restrict__ tabAi,
    float2* __restrict__ tabAin, float2* __restrict__ tabT1,
    float2* __restrict__ tabW32) {
  const int t = (int)threadIdx.x;
  for (int i = t; i < TABA_N; i += 256) {           // [d][lane][chunk]
    int d = i >> 7, lane = (i >> 2) & 31, c = i & 3;
    float dir = d ? 1.0f : -1.0f;
    int lhalf = lane >> 4, m = lane & 15;
    int kb = 4 * c + 2 * lhalf;
    float s0, c0, s1, c1;
    __sincosf(dir * TWO_PI * (float)(m * kb) * (1.0f / 16.0f), &s0, &c0);
    __sincosf(dir * TWO_PI * (float)(m * (kb + 1)) * (1.0f / 16.0f), &s1, &c1);
    tabAr[i]  = make_float2(c0, c1);
    tabAi[i]  = make_float2(s0, s1);
    tabAin[i] = make_float2(-s0, -s1);
  }
  for (int i = t; i < TABT1_N; i += 256) {          // [d][k1*32+n2]
    int d = i >> 9, k1 = (i >> 5) & 15, n2 = i & 31;
    float dir = d ? 1.0f : -1.0f;
    float s, c;
    __sincosf(dir * TWO_PI * (float)(k1 * n2) * (1.0f / 512.0f), &s, &c);
    tabT1[i] = make_float2(c, s);
  }
  for (int i = t; i < TABW32_N; i += 256) {         // [d][j1]
    int d = i >> 4, j1 = i & 15;
    float dir = d ? 1.0f : -1.0f;
    float s, c;
    __sincosf(dir * TWO_PI * (float)j1 * (1.0f / 32.0f), &s, &c);
    tabW32[i] = make_float2(c, s);
  }
}

// ---------------------------------------------------------------------------
// Angular spectrum transfer function H (512x512 complex), L2-resident.
__global__ __launch_bounds__(256) void msp_init_H(float2* __restrict__ H) {
  int idx = blockIdx.x * 256 + threadIdx.x;     // 0 .. 262143
  int iy = idx >> 9, ix = idx & 511;
  float fy = (float)(iy < 256 ? iy : iy - 512) * (1.0f / 51.2f);  // fftfreq(512,0.1)
  float fx = (float)(ix < 256 ? ix : ix - 512) * (1.0f / 51.2f);
  float fz2 = 400.0f - fy * fy - fx * fx;       // (1/0.05)^2 - fy^2 - fx^2
  float2 h = make_float2(0.0f, 0.0f);
  if (fz2 >= 0.0f) {
    float phase = TWO_PI * 0.1f * sqrtf(fz2);   // 2*pi*dz*fz
    __sincosf(phase, &h.y, &h.x);               // exp(i*phase)
  }
  H[idx] = h;
}

// ---------------------------------------------------------------------------
// Row pass: per wave one row.
//   mode 0: field = obj[:,:,0];            FFT_x        -> dst
//   mode 1: iFFT_x(src); field *= obj_z;   FFT_x        -> dst
//   mode 2: iFFT_x(src);                   re/im planes -> out
__global__ __launch_bounds__(256) void msp_row_pass(
    float2* __restrict__ dst, const float2* __restrict__ src,
    const float* __restrict__ objR, const float* __restrict__ objI,
    const float2* __restrict__ tabAr, const float2* __restrict__ tabAi,
    const float2* __restrict__ tabAin, const float2* __restrict__ tabT1,
    const float2* __restrict__ tabW32,
    int z, int mode, float* __restrict__ out) {
  __shared__ float rowsR[8][N1D];
  __shared__ float rowsI[8][N1D];
  __shared__ float scrR[8][N1D];
  __shared__ float scrI[8][N1D];
  __shared__ float2 sAr[TABA_N];
  __shared__ float2 sAi[TABA_N];
  __shared__ float2 sAin[TABA_N];
  __shared__ float2 sT1[TABT1_N];
  __shared__ float2 sW32[TABW32_N];
  const unsigned w = threadIdx.x >> 5, lane = threadIdx.x & 31;
  const int y = blockIdx.x * 8 + (int)w;
  float* bR = rowsR[w];
  float* bI = rowsI[w];

  load_tables(sAr, sAi, sAin, sT1, sW32, tabAr, tabAi, tabAin, tabT1, tabW32);
  __syncthreads();

  if (mode == 0) {
    for (int x = (int)lane; x < N1D; x += 32) {
      size_t o = ((size_t)y * N1D + x) * NZ + z;
      int p = permi(x);
      bR[p] = objR[o];
      bI[p] = objI[o];
    }
    fft512_wave(bR, bI, scrR[w], scrI[w], &sAr[0], &sAi[0], &sAin[0],
                &sT1[0], &sW32[0], 1.0f, 0, lane);               // forward FFT_x
  } else {
    for (int x = (int)lane; x < N1D; x += 32) {
      float2 v = src[(size_t)y * N1D + x];
      int p = permi(x);
      bR[p] = v.x; bI[p] = v.y;
    }
    if (mode == 1) {
      // Prefetch the strided obj slice while the inverse FFT runs; lines are
      // reused across 32 consecutive z-slices out of L2 (64 MB window).
      for (int x = (int)lane; x < N1D; x += 32) {
        size_t o = ((size_t)y * N1D + x) * NZ + z;
        __builtin_prefetch(&objR[o], 0, 0);
        __builtin_prefetch(&objI[o], 0, 0);
      }
      // iFFT keeps permuted layout so multiply is in-place per address and
      // the forward FFT consumes it directly.
      fft512_wave(bR, bI, scrR[w], scrI[w], &sAr[128], &sAi[128], &sAin[128],
                  &sT1[512], &sW32[16], 1.0f / 512.0f, 1, lane); // inverse FFT_x
      for (int x = (int)lane; x < N1D; x += 32) {
        size_t o = ((size_t)y * N1D + x) * NZ + z;
        int p = permi(x);
        float fr = bR[p], fi = bI[p];
        float orl = objR[o], oim = objI[o];
        bR[p] = fr * orl - fi * oim;
        bI[p] = fr * oim + fi * orl;
      }
      fft512_wave(bR, bI, scrR[w], scrI[w], &sAr[0], &sAi[0], &sAin[0],
                  &sT1[0], &sW32[0], 1.0f, 0, lane);             // forward FFT_x
    } else {
      fft512_wave(bR, bI, scrR[w], scrI[w], &sAr[128], &sAi[128], &sAin[128],
                  &sT1[512], &sW32[16], 1.0f / 512.0f, 0, lane); // inverse FFT_x
    }
  }

  if (mode == 2) {
    for (int x = (int)lane; x < N1D; x += 32) {
      out[(size_t)y * N1D + x] = bR[x];                          // real plane
      out[(size_t)(N1D * N1D) + (size_t)y * N1D + x] = bI[x];    // imag plane
    }
  } else {
    for (int x = (int)lane; x < N1D; x += 32)
      dst[(size_t)y * N1D + x] = make_float2(bR[x], bI[x]);
  }
}

// ---------------------------------------------------------------------------
// Column pass: FFT_y -> *H -> iFFT_y. 8 columns per block (one per wave);
// tile is column-contiguous (pitch 520 floats for bank spread) so the FFT
// works stride-1; global traffic stays row-contiguous via cooperative fill.
__global__ __launch_bounds__(256) void msp_col_pass(
    float2* __restrict__ dst, const float2* __restrict__ src,
    const float2* __restrict__ H,
    const float2* __restrict__ tabAr, const float2* __restrict__ tabAi,
    const float2* __restrict__ tabAin, const float2* __restrict__ tabT1,
    const float2* __restrict__ tabW32) {
  __shared__ float tileR[8 * CPITCH];
  __shared__ float tileI[8 * CPITCH];
  __shared__ float scrR[8][N1D];
  __shared__ float scrI[8][N1D];
  __shared__ float2 sAr[TABA_N];
  __shared__ float2 sAi[TABA_N];
  __shared__ float2 sAin[TABA_N];
  __shared__ float2 sT1[TABT1_N];
  __shared__ float2 sW32[TABW32_N];
  const unsigned w = threadIdx.x >> 5, lane = threadIdx.x & 31;
  const int kxb = blockIdx.x * 8;

  load_tables(sAr, sAi, sAin, sT1, sW32, tabAr, tabAi, tabAin, tabT1, tabW32);
  for (int i = (int)threadIdx.x; i < N1D * 8; i += 256) {
    int yy = i >> 3, c = i & 7;                 // row-contiguous global reads
    float2 v = src[(size_t)yy * N1D + kxb + c];
    int p = c * CPITCH + permi(yy);
    tileR[p] = v.x; tileI[p] = v.y;
  }
  __syncthreads();

  const int c = (int)w;
  float* bR = &tileR[c * CPITCH];
  float* bI = &tileI[c * CPITCH];
  fft512_wave(bR, bI, scrR[w], scrI[w], &sAr[0], &sAi[0], &sAin[0],
              &sT1[0], &sW32[0], 1.0f, 1, lane);                 // FFT_y (perm out)
  const int kx = kxb + c;
  for (int yy = (int)lane; yy < N1D; yy += 32) {
    float2 h = H[(size_t)yy * N1D + kx];
    int p = permi(yy);
    float fr = bR[p], fi = bI[p];
    bR[p] = fr * h.x - fi * h.y;
    bI[p] = fr * h.y + fi * h.x;
  }
  fft512_wave(bR, bI, scrR[w], scrI[w], &sAr[128], &sAi[128], &sAin[128],
              &sT1[512], &sW32[16], 1.0f / 512.0f, 0, lane);     // iFFT_y (natural)

  __syncthreads();
  for (int i = (int)threadIdx.x; i < N1D * 8; i += 256) {
    int yy = i >> 3, c2 = i & 7;
    dst[(size_t)yy * N1D + kxb + c2] =
        make_float2(tileR[c2 * CPITCH + yy], tileI[c2 * CPITCH + yy]);
  }
}

// ---------------------------------------------------------------------------
extern "C" void kernel_launch(void* const* d_in, const int* in_sizes, int n_in,
                              void* d_out, int out_size, void* d_ws, size_t ws_size,
                              hipStream_t stream) {
  const float* objR = (const float*)d_in[0];
  const float* objI = (const float*)d_in[1];
  float* out = (float*)d_out;

  // Workspace: H (2 MB) + two complex ping-pong buffers (2 MB each) + tables.
  float2* H    = (float2*)d_ws;
  float2* bufA = H + (size_t)N1D * N1D;
  float2* bufB = bufA + (size_t)N1D * N1D;
  float2* tabAr  = bufB + (size_t)N1D * N1D;       // 2 KB
  float2* tabAi  = tabAr + TABA_N;                 // 2 KB
  float2* tabAin = tabAi + TABA_N;                 // 2 KB
  float2* tabT1  = tabAin + TABA_N;                // 8 KB
  float2* tabW32 = tabT1 + TABT1_N;                // 256 B

  msp_init_tw<<<1, 256, 0, stream>>>(tabAr, tabAi, tabAin, tabT1, tabW32);
  msp_init_H<<<(N1D * N1D) / 256, 256, 0, stream>>>(H);

  // field0 spectrum-in-x: FFT_x(obj[:,:,0])
  msp_row_pass<<<N1D / 8, 256, 0, stream>>>(bufA, bufA, objR, objI,
                                            tabAr, tabAi, tabAin, tabT1, tabW32,
                                            0, 0, out);

  // 255 multiply+propagate steps
  for (int s = 1; s < NZ; ++s) {
    msp_col_pass<<<N1D / 8, 256, 0, stream>>>(bufB, bufA, H,
                                              tabAr, tabAi, tabAin, tabT1, tabW32);
    msp_row_pass<<<N1D / 8, 256, 0, stream>>>(bufA, bufB, objR, objI,
                                              tabAr, tabAi, tabAin, tabT1, tabW32,
                                              s, 1, out);
  }

  // final propagation of the last multiplied field, then iFFT_x -> output
  msp_col_pass<<<N1D / 8, 256, 0, stream>>>(bufB, bufA, H,
                                            tabAr, tabAi, tabAin, tabT1, tabW32);
  msp_row_pass<<<N1D / 8, 256, 0, stream>>>(bufA, bufB, objR, objI,
                                            tabAr, tabAi, tabAin, tabT1, tabW32,
                                            0, 2, out);
}